// BaseLikelihoodModel_27058293965144
// MI455X (gfx1250) — compile-verified
//
#include <hip/hip_runtime.h>
#include <hip/hip_bf16.h>
#include <math.h>

typedef __attribute__((ext_vector_type(2))) float v2f;
typedef __attribute__((ext_vector_type(8))) float v8f;

#define SDIM 64
#define LDA  65          // LDS row stride (pad to avoid 64-bank conflicts)
#define EPSF 1e-30f
#define TPB  512

// One 16x16 C-tile of D = A(64x64) x B(64x64), both row-major in LDS with stride LDA.
// K swept in 16 chunks of 4 via V_WMMA_F32_16X16X4_F32.
__device__ __forceinline__ v8f mm_tile(const float* __restrict__ A,
                                       const float* __restrict__ B,
                                       int ti, int tj, int lane) {
  const int m  = lane & 15;        // row (A) / col (B) within tile
  const int kh = (lane >> 4) << 1; // 0 or 2: which K-pair this half-wave supplies
  v8f acc = {};
#pragma unroll
  for (int kb = 0; kb < 16; ++kb) {
    const int k0 = kb * 4 + kh;
    v2f av, bv;
    av.x = A[(ti * 16 + m) * LDA + k0];
    av.y = A[(ti * 16 + m) * LDA + k0 + 1];
    bv.x = B[(k0)     * LDA + tj * 16 + m];
    bv.y = B[(k0 + 1) * LDA + tj * 16 + m];
    acc = __builtin_amdgcn_wmma_f32_16x16x4_f32(false, av, false, bv,
                                                (short)0, acc, false, false);
  }
  return acc;
}

// expm((Q - diag(g)) * t) -> Pm (row-major, stride LDA). Uses Bm as scratch.
// Scaling-and-squaring + degree-8 Taylor (Horner). t==0 -> identity naturally.
__device__ void expm_lds(float* __restrict__ Bm, float* __restrict__ Pm,
                         float* __restrict__ rs,
                         const float* __restrict__ Q,
                         const float* __restrict__ growth,
                         float t, int tid) {
  // A = (Q - diag(growth)) * t
  for (int idx = tid; idx < SDIM * SDIM; idx += TPB) {
    const int r = idx >> 6, c = idx & 63;
    const float q = Q[idx] - ((r == c) ? growth[r] : 0.0f);
    Bm[r * LDA + c] = q * t;
  }
  __syncthreads();
  // inf-norm
  if (tid < SDIM) {
    float s = 0.0f;
    for (int j = 0; j < SDIM; ++j) s += fabsf(Bm[tid * LDA + j]);
    rs[tid] = s;
  }
  __syncthreads();
  if (tid == 0) {
    float nrm = 0.0f;
    for (int j = 0; j < SDIM; ++j) nrm = fmaxf(nrm, rs[j]);
    int sc = 0;
    if (nrm > 0.25f) {
      sc = (int)ceilf(log2f(nrm * 4.0f));
      if (sc < 1)  sc = 1;
      if (sc > 30) sc = 30;
    }
    rs[64] = (float)sc;
    rs[65] = ldexpf(1.0f, -sc);
  }
  __syncthreads();
  const int   sc  = (int)rs[64];
  const float scl = rs[65];
  constexpr int K = 8;
  // scale A, init P = I + A/K  (first Horner step folded in; same-thread idx mapping)
  for (int idx = tid; idx < SDIM * SDIM; idx += TPB) {
    const int r = idx >> 6, c = idx & 63;
    const float b = Bm[r * LDA + c] * scl;
    Bm[r * LDA + c] = b;
    Pm[r * LDA + c] = ((r == c) ? 1.0f : 0.0f) + b * (1.0f / (float)K);
  }
  __syncthreads();

  const int lane = tid & 31, wave = tid >> 5;
  const int ti = wave >> 2, tj = wave & 3;
  const int m = lane & 15, khalf = lane >> 4;

  // Horner: P = I + (B*P)/k, k = K-1 .. 1
  for (int k = K - 1; k >= 1; --k) {
    v8f acc = mm_tile(Bm, Pm, ti, tj, lane);
    __syncthreads();
    const float invk = 1.0f / (float)k;
#pragma unroll
    for (int r = 0; r < 8; ++r) {
      const int row = ti * 16 + khalf * 8 + r;
      const int col = tj * 16 + m;
      Pm[row * LDA + col] = ((row == col) ? 1.0f : 0.0f) + acc[r] * invk;
    }
    __syncthreads();
  }
  // squaring
  for (int it = 0; it < sc; ++it) {
    v8f acc = mm_tile(Pm, Pm, ti, tj, lane);
    __syncthreads();
#pragma unroll
    for (int r = 0; r < 8; ++r) {
      const int row = ti * 16 + khalf * 8 + r;
      const int col = tj * 16 + m;
      Pm[row * LDA + col] = acc[r];
    }
    __syncthreads();
  }
}

// One block per tree node at `lev`: expm for each child, then
// lp[node][i] = logsumexp_j(logT_L[i,j]+lp[L][j]) + [logsumexp_j(logT_R[i,j]+lp[R][j]) + log g[i]]
__global__ __launch_bounds__(TPB)
void level_kernel(const int* __restrict__ level_nodes,
                  const int* __restrict__ children,
                  const float* __restrict__ blens,
                  const float* __restrict__ Q,
                  const float* __restrict__ growth,
                  float* __restrict__ lp,
                  int lev, int W, int N) {
  __shared__ float Bm[SDIM * LDA];
  __shared__ float Pm[SDIM * LDA];
  __shared__ float rs[66];
  __shared__ float vbuf[SDIM];
  __shared__ float accv[SDIM];

  const int tid  = threadIdx.x;
  const int node = level_nodes[(size_t)lev * W + blockIdx.x];
  if (node < 0 || node >= N) return;   // padding slot

  const int left  = children[node * 2 + 0];
  const int right = children[node * 2 + 1];

  if (tid < SDIM) accv[tid] = 0.0f;

  for (int side = 0; side < 2; ++side) {
    const int child = (side == 0) ? left : right;
    if (child < 0) break;              // root has no right child (uniform per block)

    expm_lds(Bm, Pm, rs, Q, growth, blens[child], tid);

    // Bm <- log(max(P, EPS));  vbuf <- lp[child]
    for (int idx = tid; idx < SDIM * SDIM; idx += TPB) {
      const int r = idx >> 6, c = idx & 63;
      Bm[r * LDA + c] = logf(fmaxf(Pm[r * LDA + c], EPSF));
    }
    if (tid < SDIM) vbuf[tid] = lp[(size_t)child * SDIM + tid];
    __syncthreads();

    if (tid < SDIM) {
      float mx = -3.4e38f;
      for (int j = 0; j < SDIM; ++j)
        mx = fmaxf(mx, Bm[tid * LDA + j] + vbuf[j]);
      float sum = 0.0f;
      for (int j = 0; j < SDIM; ++j)
        sum += expf(Bm[tid * LDA + j] + vbuf[j] - mx);
      float r = mx + logf(sum);
      if (side == 1) r += logf(growth[tid]);   // right child carries log-growth term
      accv[tid] += r;
    }
    __syncthreads();   // protect Bm/Pm reuse for the next child
  }

  if (tid < SDIM) lp[(size_t)node * SDIM + tid] = accv[tid];
}

__global__ void init_lp_kernel(const float* __restrict__ init_partials,
                               float* __restrict__ lp, int n) {
  const int i = blockIdx.x * 256 + threadIdx.x;
  if (i < n) lp[i] = init_partials[i];
}

__global__ void write_out_kernel(const float* __restrict__ lp,
                                 float* __restrict__ out, int root) {
  out[threadIdx.x] = lp[(size_t)root * SDIM + threadIdx.x];
}

extern "C" void kernel_launch(void* const* d_in, const int* in_sizes, int n_in,
                              void* d_out, int out_size, void* d_ws, size_t ws_size,
                              hipStream_t stream) {
  const int*   children = (const int*)  d_in[1];
  const float* blens    = (const float*)d_in[2];
  const float* init_p   = (const float*)d_in[3];
  const float* Q        = (const float*)d_in[4];
  const float* growth   = (const float*)d_in[6];
  const int*   lvlnodes = (const int*)  d_in[7];

  const int N = in_sizes[2];       // 2L nodes
  const int L = N / 2;
  const int W = L / 2;
  const int num_levels = in_sizes[7] / W;
  const int root = N - 1;

  float* lp = (float*)d_ws;        // (N x 64) floats

  // lp <- init_partials
  {
    const int n = N * SDIM;
    init_lp_kernel<<<(n + 255) / 256, 256, 0, stream>>>(init_p, lp, n);
  }

  // level-by-level pruning (each level depends on the previous; stream order)
  for (int lev = 0; lev < num_levels; ++lev) {
    int nb = L >> (lev + 1);
    if (nb < 1) nb = 1;
    if (nb > W) nb = W;
    level_kernel<<<nb, TPB, 0, stream>>>(lvlnodes, children, blens, Q, growth,
                                         lp, lev, W, N);
  }

  write_out_kernel<<<1, SDIM, 0, stream>>>(lp, (float*)d_out, root);
}